// LGA_41188736369106
// MI455X (gfx1250) — compile-verified
//
#include <hip/hip_runtime.h>
#include <cstdint>
#include <cstddef>

// ---------------------------------------------------------------------------
// Problem constants (from reference): B=8, G=1024, K=32, C=168, OUT_DIM=336
// out[b, c, g, k] = (knn_x_w + pe) * pe,  shape [8, 336, 1024, 32] fp32
// ---------------------------------------------------------------------------
#define B_    8
#define G_    1024
#define K_    32
#define C_    168
#define OUTD  336
#define BLK   256
#define RED_BLOCKS 512

// -log2(1000)/56  (dim_embed[f] = 1000^(f/56); arg = 100*t*1000^(-f/56))
#define NEGK (-0.1779604336546801f)

typedef int v2i __attribute__((ext_vector_type(2)));
typedef __attribute__((address_space(1))) v2i gv2i_t;  // global b64 chunk
typedef __attribute__((address_space(3))) v2i lv2i_t;  // LDS b64 chunk

#if __has_builtin(__builtin_amdgcn_global_load_async_to_lds_b64)
#define HAVE_ASYNC_LDS 1
#else
#define HAVE_ASYNC_LDS 0
#endif

__device__ __forceinline__ float fast_exp2(float x) {
#if __has_builtin(__builtin_amdgcn_exp2f)
  return __builtin_amdgcn_exp2f(x);
#else
  return exp2f(x);
#endif
}

__device__ __forceinline__ void wait_async_zero() {
#if __has_builtin(__builtin_amdgcn_s_wait_asynccnt)
  __builtin_amdgcn_s_wait_asynccnt(0);
#else
  asm volatile("s_wait_asynccnt 0" ::: "memory");
#endif
}

__device__ __forceinline__ void stage_b64(const float* gp, float* lp) {
#if HAVE_ASYNC_LDS
  __builtin_amdgcn_global_load_async_to_lds_b64(
      (gv2i_t*)(uintptr_t)gp, (lv2i_t*)(uint32_t)(uintptr_t)lp, 0, 0);
#else
  *reinterpret_cast<float2*>(lp) = *reinterpret_cast<const float2*>(gp);
#endif
}

// ---------------------------------------------------------------------------
// Main fused kernel. One block per (b,g). 8 waves, lane == k.
//   - async-stage knn_x[bg] tile (32x168 f32) into padded LDS (row stride 170
//     floats -> transposed reads are bank conflict free: 42k mod 64 distinct).
//     Staging is division-free: wave -> row, lane -> b64 chunk (12 unrolled
//     global_load_async_to_lds_b64, tail masked via EXEC).
//   - precompute pe_lc[336] once per block (hoists half the transcendentals)
//   - each wave sweeps 42 channels; one v_sin/v_cos per output element;
//     stores are 128 B coalesced non-temporal rows of out[b,c,g,:]
// ---------------------------------------------------------------------------
__global__ void __launch_bounds__(BLK)
LGA_main(const float* __restrict__ lc_xyz, const float* __restrict__ lc_x,
         const float* __restrict__ knn_xyz, const float* __restrict__ knn_x,
         const double* __restrict__ ws, float* __restrict__ out) {
  constexpr int KROW = C_ + 2;            // 170 floats, 8B-aligned rows
  __shared__ float s_knnx[K_ * KROW];     // 21760 B
  __shared__ float s_lcx[C_];
  __shared__ float s_emb[K_ * 9];         // [k][7], stride 9 (conflict free)
  __shared__ float s_pelc[OUTD];

  const int tid = threadIdx.x;
  const int bg  = blockIdx.x;
  const int wv  = tid >> 5;               // wave 0..7
  const int k   = tid & 31;               // lane == neighbor index

  // ---- FIRST: kick off async knn_x tile -> LDS (CDNA5 ASYNCcnt path) ----
  {
    const float* gsrc = knn_x + (size_t)bg * (K_ * C_);
#pragma unroll
    for (int rr = 0; rr < 4; ++rr) {            // wave handles rows wv+8*rr
      const int r = wv + rr * 8;
      const float* grow = gsrc + r * C_;
      float*       lrow = s_knnx + r * KROW;
#pragma unroll
      for (int u = 0; u < 3; ++u) {             // 84 b64 chunks per row
        const int cc = k + u * 32;
        if (u < 2 || cc < C_ / 2) {             // tail: lanes < 20
          stage_b64(grow + cc * 2, lrow + cc * 2);
        }
      }
    }
  }

  const int b = bg >> 10;                 // G_ = 1024
  const int g = bg & (G_ - 1);

  // global stds (jnp.std, ddof=1): var = (sumsq - sum^2/N) / (N-1)
  const double Nx = (double)B_ * G_ * K_ * C_;
  const double Nz = (double)B_ * G_ * K_ * 3;
  const double sx = ws[0], qx = ws[1], sz = ws[2], qz = ws[3];
  const float inv_sx = 1.0f / ((float)sqrt((qx - sx * sx / Nx) / (Nx - 1.0)) + 1e-5f);
  const float inv_sz = 1.0f / ((float)sqrt((qz - sz * sz / Nz) / (Nz - 1.0)) + 1e-5f);

  // lc_x row -> LDS
  for (int c = tid; c < C_; c += BLK) s_lcx[c] = lc_x[(size_t)bg * C_ + c];

  // lc_xyz (wave-uniform scalars)
  const float lx = lc_xyz[bg * 3 + 0];
  const float ly = lc_xyz[bg * 3 + 1];
  const float lz = lc_xyz[bg * 3 + 2];

  // per-lane neighbor xyz, normalized; cross & dot vs (raw) lc
  const float* kp = knn_xyz + ((size_t)bg * K_ + k) * 3;
  const float xn = (kp[0] - lx) * inv_sz;
  const float yn = (kp[1] - ly) * inv_sz;
  const float zn = (kp[2] - lz) * inv_sz;
  const float cx = yn * lz - zn * ly;
  const float cy = zn * lx - xn * lz;
  const float cz = xn * ly - yn * lx;
  const float dt = xn * lx + yn * ly + zn * lz;
  if (wv == 0) {
    s_emb[k * 9 + 0] = xn; s_emb[k * 9 + 1] = yn; s_emb[k * 9 + 2] = zn;
    s_emb[k * 9 + 3] = cx; s_emb[k * 9 + 4] = cy; s_emb[k * 9 + 5] = cz;
    s_emb[k * 9 + 6] = dt;
  }

  // positional embedding of lc (channel-only) hoisted: 336 trig per block
  for (int c = tid; c < OUTD; c += BLK) {
    const int dim = c / 112;
    const int rem = c - dim * 112;
    const int sc  = rem / 56;
    const int f   = rem - sc * 56;
    const float wf = 100.0f * fast_exp2((float)f * NEGK);
    const float t  = (dim == 0) ? lx : ((dim == 1) ? ly : lz);
    const float a  = t * wf;
    s_pelc[c] = sc ? __cosf(a) : __sinf(a);
  }

#if HAVE_ASYNC_LDS
  wait_async_zero();
#endif
  __syncthreads();

  // ---- main sweep: wave wv owns channels [wv*42, wv*42+42) ----
  const size_t outbase = (size_t)b * OUTD * (G_ * K_) + (size_t)g * K_ + k;
  for (int it = 0; it < OUTD / 8; ++it) {            // 42
    const int c   = wv * (OUTD / 8) + it;
    const int dim = c / 112;
    const int rem = c - dim * 112;
    const int sc  = rem / 56;
    const int f   = rem - sc * 56;
    const float wf = 100.0f * fast_exp2((float)f * NEGK);
    const float tk = (dim == 0) ? xn : ((dim == 1) ? yn : zn);
    const float a  = tk * wf;
    const float pe = (sc ? __cosf(a) : __sinf(a)) + s_pelc[c];

    float xw;
    if (c < C_) {
      xw = (s_knnx[k * KROW + c] - s_lcx[c]) * inv_sx;   // normalized knn_x
    } else {
      const int j = (c - C_) % 7;                         // tiled 7-ch embedding
      xw = s_emb[k * 9 + j];
    }
    // streamed write-once output: non-temporal, keep L2 for the inputs
    __builtin_nontemporal_store((xw + pe) * pe, &out[outbase + (size_t)c * (G_ * K_)]);
  }
}

// ---------------------------------------------------------------------------
// Pass 1: per-block partial sums / sums-of-squares of
//   d_x   = knn_x  - lc_x  (broadcast over K)   (44,040,192 elems)
//   d_xyz = knn_xyz - lc_xyz (broadcast over K) (   786,432 elems)
// Deterministic: fixed grid-stride ranges, no atomics. All 32-bit indexing
// (N4 < 2^31) -> no v_mul_u64 magic-division in the hot loop.
// ws layout (doubles): [0..3] finals, [8 + 4*blk .. ] partials.
// ---------------------------------------------------------------------------
__global__ void __launch_bounds__(BLK)
LGA_reduce1(const float* __restrict__ knn_x, const float* __restrict__ lc_x,
            const float* __restrict__ knn_xyz, const float* __restrict__ lc_xyz,
            double* __restrict__ ws) {
  const uint32_t tid  = threadIdx.x;
  const uint32_t gtid = blockIdx.x * BLK + tid;
  const uint32_t gstr = gridDim.x * BLK;

  double s0 = 0.0, q0 = 0.0, s1 = 0.0, q1 = 0.0;

  // knn_x part, vectorized float4 (C=168 divisible by 4 -> no row crossing)
  constexpr uint32_t ROW4 = C_ / 4;             // 42 float4 per row
  constexpr uint32_t BG4  = K_ * C_ / 4;        // 1344 float4 per (b,g)
  const uint32_t N4 = (uint32_t)B_ * G_ * BG4;  // 11,010,048
  const float4* kx4 = reinterpret_cast<const float4*>(knn_x);
  for (uint32_t i = gtid; i < N4; i += gstr) {
    const uint32_t bg = i / BG4;                // 32-bit magic mul
    const uint32_t j  = i - bg * BG4;
    const uint32_t c4 = (j % ROW4) * 4;         // channel offset within row
    float4 a = kx4[i];
    float4 r = *reinterpret_cast<const float4*>(lc_x + (size_t)bg * C_ + c4);
    float d0 = a.x - r.x, d1 = a.y - r.y, d2 = a.z - r.z, d3 = a.w - r.w;
    s0 += (double)d0 + (double)d1 + (double)d2 + (double)d3;
    q0 += (double)d0 * d0 + (double)d1 * d1 + (double)d2 * d2 + (double)d3 * d3;
  }

  // knn_xyz part (small)
  const uint32_t Nz = (uint32_t)B_ * G_ * K_ * 3;  // 786,432
  for (uint32_t i = gtid; i < Nz; i += gstr) {
    const uint32_t c  = i % 3u;
    const uint32_t bg = i / (K_ * 3u);
    float d = knn_xyz[i] - lc_xyz[bg * 3 + c];
    s1 += (double)d;
    q1 += (double)d * d;
  }

  // wave32 tree reduce then cross-wave via LDS
  for (int off = 16; off > 0; off >>= 1) {
    s0 += __shfl_down(s0, off, 32);
    q0 += __shfl_down(q0, off, 32);
    s1 += __shfl_down(s1, off, 32);
    q1 += __shfl_down(q1, off, 32);
  }
  __shared__ double red[(BLK / 32) * 4];
  const uint32_t wave = tid >> 5, lane = tid & 31;
  if (lane == 0) {
    red[wave * 4 + 0] = s0; red[wave * 4 + 1] = q0;
    red[wave * 4 + 2] = s1; red[wave * 4 + 3] = q1;
  }
  __syncthreads();
  if (tid == 0) {
    double t0 = 0, t1 = 0, t2 = 0, t3 = 0;
    for (int w = 0; w < BLK / 32; ++w) {
      t0 += red[w * 4 + 0]; t1 += red[w * 4 + 1];
      t2 += red[w * 4 + 2]; t3 += red[w * 4 + 3];
    }
    double* p = ws + 8 + (size_t)blockIdx.x * 4;
    p[0] = t0; p[1] = t1; p[2] = t2; p[3] = t3;
  }
}

// ---------------------------------------------------------------------------
// Pass 2: fixed-order reduction of the RED_BLOCKS partials -> ws[0..3]
// ---------------------------------------------------------------------------
__global__ void __launch_bounds__(BLK)
LGA_reduce2(double* __restrict__ ws) {
  const int tid = threadIdx.x;
  double t0 = 0, t1 = 0, t2 = 0, t3 = 0;
  for (int bk = tid; bk < RED_BLOCKS; bk += BLK) {
    const double* p = ws + 8 + (size_t)bk * 4;
    t0 += p[0]; t1 += p[1]; t2 += p[2]; t3 += p[3];
  }
  for (int off = 16; off > 0; off >>= 1) {
    t0 += __shfl_down(t0, off, 32);
    t1 += __shfl_down(t1, off, 32);
    t2 += __shfl_down(t2, off, 32);
    t3 += __shfl_down(t3, off, 32);
  }
  __shared__ double red[(BLK / 32) * 4];
  const int wave = tid >> 5, lane = tid & 31;
  if (lane == 0) {
    red[wave * 4 + 0] = t0; red[wave * 4 + 1] = t1;
    red[wave * 4 + 2] = t2; red[wave * 4 + 3] = t3;
  }
  __syncthreads();
  if (tid == 0) {
    double a0 = 0, a1 = 0, a2 = 0, a3 = 0;
    for (int w = 0; w < BLK / 32; ++w) {
      a0 += red[w * 4 + 0]; a1 += red[w * 4 + 1];
      a2 += red[w * 4 + 2]; a3 += red[w * 4 + 3];
    }
    ws[0] = a0; ws[1] = a1; ws[2] = a2; ws[3] = a3;
  }
}

// ---------------------------------------------------------------------------
extern "C" void kernel_launch(void* const* d_in, const int* in_sizes, int n_in,
                              void* d_out, int out_size, void* d_ws, size_t ws_size,
                              hipStream_t stream) {
  (void)in_sizes; (void)n_in; (void)out_size; (void)ws_size;
  const float* lc_xyz  = (const float*)d_in[0];
  const float* lc_x    = (const float*)d_in[1];
  const float* knn_xyz = (const float*)d_in[2];
  const float* knn_x   = (const float*)d_in[3];
  double* ws  = (double*)d_ws;   // uses 8 + RED_BLOCKS*4 doubles = 16,448 B
  float*  out = (float*)d_out;

  LGA_reduce1<<<RED_BLOCKS, BLK, 0, stream>>>(knn_x, lc_x, knn_xyz, lc_xyz, ws);
  LGA_reduce2<<<1, BLK, 0, stream>>>(ws);
  LGA_main<<<B_ * G_, BLK, 0, stream>>>(lc_xyz, lc_x, knn_xyz, knn_x, ws, out);
}